// GIP_Generator_33105607918293
// MI455X (gfx1250) — compile-verified
//
#include <hip/hip_runtime.h>
#include <math.h>

typedef float v2f __attribute__((ext_vector_type(2)));
typedef float v8f __attribute__((ext_vector_type(8)));

#define NB 384      // haplotypes (rows)
#define NF 1024     // features
#define NK1 2048    // K of first GEMM (concat x|m)
#define GAMMA_F 1.0e4f

// ---------------------------------------------------------------------------
// WMMA fp32 GEMM: out[M,N] = act(A @ W^T + bias [+ extra])
//   A: [M,K] row-major (CONCAT: the K axis is x|m, each segment [M, NF])
//   W: [N,K] row-major  (B-matrix B[k][n] = W[n][k]; rows contiguous in K)
// Each wave computes a 16x32 output tile: two 16x16 WMMA accumulators that
// share one A fragment per K-step (1.5 b64 loads / wmma, two independent
// accumulation chains for the XDL pipeline). Block = 8 waves -> 32x128 tile.
// Fragment layout per CDNA5 ISA 7.12.2:
//   A 16x4:  lanes 0-15 hold (M=lane, K=0,1), lanes 16-31 hold (M=lane-16, K=2,3)
//   B 4x16:  lanes 0-15 hold (K=0,1, N=lane), lanes 16-31 hold (K=2,3, N=lane-16)
//   C/D:     VGPR v: row = v + 8*half, col = lane&15
// The concat split is handled as two sequential K segments: no per-iteration
// pointer select (kills the v_cndmask chains seen in round 1).
// ---------------------------------------------------------------------------
template<bool CONCAT, bool RELU, bool ADD_EXTRA>
__global__ __launch_bounds__(256)
void gemm_wmma_kernel(const float* __restrict__ A0,
                      const float* __restrict__ A1,
                      const float* __restrict__ W,
                      const float* __restrict__ bias,
                      const float* __restrict__ extra,
                      float* __restrict__ out,
                      int M, int N, int K)
{
    const int lane = threadIdx.x & 31;
    const int wave = threadIdx.x >> 5;     // 8 waves per block
    const int wm   = wave & 1;             // 2 waves along M
    const int wn   = wave >> 1;            // 4 waves along N (32 cols each)
    const int m_base = blockIdx.y * 32 + wm * 16;
    const int n_base = blockIdx.x * 128 + wn * 32;
    const int r    = lane & 15;
    const int half = lane >> 4;
    const int koff = 2 * half;

    const int a_row = m_base + r;
    const int lda   = CONCAT ? NF : K;
    const float* __restrict__ w_row0 = W + (size_t)(n_base + r) * K;
    const float* __restrict__ w_row1 = W + (size_t)(n_base + 16 + r) * K;

    v8f acc0 = {0.f, 0.f, 0.f, 0.f, 0.f, 0.f, 0.f, 0.f};
    v8f acc1 = {0.f, 0.f, 0.f, 0.f, 0.f, 0.f, 0.f, 0.f};

    const int  nseg = CONCAT ? 2 : 1;
    const int  Kseg = CONCAT ? NF : K;
    for (int seg = 0; seg < nseg; ++seg) {
        const float* __restrict__ ap =
            ((CONCAT && seg) ? A1 : A0) + (size_t)a_row * lda + koff;
        const float* __restrict__ wp0 = w_row0 + seg * NF + koff;
        const float* __restrict__ wp1 = w_row1 + seg * NF + koff;

        for (int k0 = 0; k0 < Kseg; k0 += 64) {
            // stream the two W rows: global_prefetch_b8 on gfx1250
            __builtin_prefetch(wp0 + k0 + 64, 0, 1);
            __builtin_prefetch(wp1 + k0 + 64, 0, 1);
#pragma unroll
            for (int kk = 0; kk < 64; kk += 4) {
                const int ka = k0 + kk;
                const v2f a  = *(const v2f*)(ap  + ka);
                const v2f b0 = *(const v2f*)(wp0 + ka);
                const v2f b1 = *(const v2f*)(wp1 + ka);
                acc0 = __builtin_amdgcn_wmma_f32_16x16x4_f32(
                           false, a, false, b0, (short)0, acc0, false, false);
                acc1 = __builtin_amdgcn_wmma_f32_16x16x4_f32(
                           false, a, false, b1, (short)0, acc1, false, false);
            }
        }
    }

    const int col0 = n_base + r;
    const int col1 = col0 + 16;
    const float bn0 = bias[col0];
    const float bn1 = bias[col1];
#pragma unroll
    for (int v = 0; v < 8; ++v) {
        const int row = m_base + v + 8 * half;
        float v0 = acc0[v] + bn0;
        float v1 = acc1[v] + bn1;
        if (ADD_EXTRA) {
            v0 += extra[(size_t)row * N + col0];
            v1 += extra[(size_t)row * N + col1];
        }
        if (RELU) {
            v0 = fmaxf(v0, 0.0f);
            v1 = fmaxf(v1, 0.0f);
        }
        out[(size_t)row * N + col0] = v0;
        out[(size_t)row * N + col1] = v1;
    }
}

// ---------------------------------------------------------------------------
// Row softmax + argmax + straight-through one-hot (forward value).
// h2[c] = (y_hard[c] - y_soft[c]) + y_soft[c]   (exact 0 off the argmax lane)
// ---------------------------------------------------------------------------
__global__ __launch_bounds__(256)
void softmax_st_kernel(const float* __restrict__ logits,
                       float* __restrict__ h2)
{
    const int row = blockIdx.x;
    const int tid = threadIdx.x;
    const float* __restrict__ lr = logits + (size_t)row * NF;

    __shared__ float sval[256];
    __shared__ int   sidx[256];

    // max + first-argmax (ties -> lowest index, matching jnp.argmax)
    float mx = -INFINITY; int mi = NF;
    for (int c = tid; c < NF; c += 256) {
        const float v = lr[c];
        if (v > mx) { mx = v; mi = c; }
    }
    sval[tid] = mx; sidx[tid] = mi;
    __syncthreads();
    for (int s = 128; s > 0; s >>= 1) {
        if (tid < s) {
            const float ov = sval[tid + s]; const int oi = sidx[tid + s];
            if (ov > sval[tid] || (ov == sval[tid] && oi < sidx[tid])) {
                sval[tid] = ov; sidx[tid] = oi;
            }
        }
        __syncthreads();
    }
    const float rowmax = sval[0];
    const int   rowarg = sidx[0];
    __syncthreads();

    // sum of exp
    float acc = 0.f;
    for (int c = tid; c < NF; c += 256) acc += expf(lr[c] - rowmax);
    sval[tid] = acc;
    __syncthreads();
    for (int s = 128; s > 0; s >>= 1) {
        if (tid < s) sval[tid] += sval[tid + s];
        __syncthreads();
    }
    const float inv_denom = 1.0f / sval[0];

    for (int c = tid; c < NF; c += 256) {
        const float ys = expf(lr[c] - rowmax) * inv_denom;
        const float yh = (c == rowarg) ? 1.0f : 0.0f;
        h2[(size_t)row * NF + c] = (yh - ys) + ys;
    }
}

// ---------------------------------------------------------------------------
// Pairwise Cauchy density: pr[i,f] = beta + aOverB * sum_j 1/(1 + g*d^2)
// h2 (1.5 MB) is L2-resident; loads are fully coalesced across threadIdx.x.
// ---------------------------------------------------------------------------
__global__ __launch_bounds__(256)
void cauchy_kernel(const float* __restrict__ z3,
                   const float* __restrict__ h2,
                   float* __restrict__ pr,
                   float alpha_over_B, float beta)
{
    const int f = blockIdx.x * 256 + threadIdx.x;
    const int i = blockIdx.y;
    const float zi = z3[(size_t)i * NF + f];
    const float* __restrict__ hcol = h2 + f;

    float s = 0.f;
#pragma unroll 4
    for (int j = 0; j < NB; ++j) {
        const float d = zi - hcol[(size_t)j * NF];
        s += 1.0f / fmaf(GAMMA_F * d, d, 1.0f);
    }
    pr[(size_t)i * NF + f] = fmaf(s, alpha_over_B, beta);
}

// ---------------------------------------------------------------------------
extern "C" void kernel_launch(void* const* d_in, const int* in_sizes, int n_in,
                              void* d_out, int out_size, void* d_ws, size_t ws_size,
                              hipStream_t stream)
{
    (void)in_sizes; (void)n_in; (void)out_size; (void)ws_size;

    const float* x      = (const float*)d_in[0];   // [384,1024]
    const float* m      = (const float*)d_in[1];   // [384,1024]
    const float* gumbel = (const float*)d_in[2];   // [384,1024]
    const float* W1     = (const float*)d_in[3];   // [1024,2048]
    const float* b1     = (const float*)d_in[4];   // [1024]
    const float* W2     = (const float*)d_in[5];   // [1024,1024]
    const float* b2     = (const float*)d_in[6];   // [1024]
    const float* W3     = (const float*)d_in[7];   // [1024,1024]
    const float* b3     = (const float*)d_in[8];   // [1024]
    float* pr = (float*)d_out;                     // [384,1024]

    const size_t act = (size_t)NB * NF;            // 393216 floats
    float* h1     = (float*)d_ws;
    float* logits = h1 + act;
    float* h2     = logits + act;
    float* z3     = h2 + act;

    // Li & Stephens mutation params, n = NB (harmonic sum over 1..n-1)
    double hs = 0.0;
    for (int k = 1; k < NB; ++k) hs += 1.0 / (double)k;
    const double theta = 1.0 / hs;
    const float alpha = (float)((double)NB / ((double)NB + theta));
    const float beta  = (float)(0.5 * theta / ((double)NB + theta));
    const float alpha_over_B = alpha / (float)NB;

    const dim3 blk(256);
    const dim3 gemm_grid(NF / 128, NB / 32);       // (8, 12)

    // h1 = relu([x|m] @ W1^T + b1)
    gemm_wmma_kernel<true, true, false><<<gemm_grid, blk, 0, stream>>>(
        x, m, W1, b1, nullptr, h1, NB, NF, NK1);

    // logits = h1 @ W2^T + b2 + gumbel
    gemm_wmma_kernel<false, false, true><<<gemm_grid, blk, 0, stream>>>(
        h1, nullptr, W2, b2, gumbel, logits, NB, NF, NF);

    // h2 = straight-through gumbel softmax (hard, forward value)
    softmax_st_kernel<<<dim3(NB), blk, 0, stream>>>(logits, h2);

    // z3 = h2 @ W3^T + b3
    gemm_wmma_kernel<false, false, false><<<gemm_grid, blk, 0, stream>>>(
        h2, nullptr, W3, b3, nullptr, z3, NB, NF, NF);

    // pr = beta + (alpha/B) * sum_j 1/(1 + 1e4*(z3[i,f]-h2[j,f])^2)
    cauchy_kernel<<<dim3(NF / 256, NB), blk, 0, stream>>>(
        z3, h2, pr, alpha_over_B, beta);
}